// Encoder_25031069401693
// MI455X (gfx1250) — compile-verified
//
#include <hip/hip_runtime.h>

typedef __attribute__((ext_vector_type(2))) float v2f;
typedef __attribute__((ext_vector_type(8))) float v8f;

#define DFEAT 128
// LDS layout for W fragments: k-pair p, column n -> sW[p*PSTR + n*2 + (k&1)]
// PSTR=288 dwords: 288%64==32 so the two lane-halves of a b64 load hit disjoint bank halves.
#define PSTR 288

// ---------------- zero ----------------
__global__ void k_zero(float* __restrict__ p, int n) {
    int i = blockIdx.x * blockDim.x + threadIdx.x;
    int stride = gridDim.x * blockDim.x;
    for (; i < n; i += stride) p[i] = 0.0f;
}

// ---------------- degree count ----------------
__global__ void k_deg(const int* __restrict__ esrc, const int* __restrict__ edst,
                      float* __restrict__ dout, float* __restrict__ din, int e) {
    int i = blockIdx.x * blockDim.x + threadIdx.x;
    if (i < e) {
        atomicAdd(&dout[esrc[i]], 1.0f);
        atomicAdd(&din[edst[i]], 1.0f);
    }
}

// deg -> rsqrt norm in place (deg is an integer count, so max(deg,1)==deg when >0)
__global__ void k_norm(float* __restrict__ d, int n) {
    int i = blockIdx.x * blockDim.x + threadIdx.x;
    if (i < n) {
        float v = d[i];
        d[i] = (v > 0.0f) ? rsqrtf(v) : 0.0f;
    }
}

// ---------------- COO scatter: agg[dst] += x[src]*nsrc[src] ----------------
// one wave per edge, float4 per lane (32*4 = 128 features)
__global__ __launch_bounds__(256) void k_scatter(const float* __restrict__ x,
                                                 const float* __restrict__ nsrc,
                                                 const int* __restrict__ esrc,
                                                 const int* __restrict__ edst,
                                                 float* __restrict__ agg, int e) {
    int gw = (blockIdx.x * blockDim.x + threadIdx.x) >> 5;
    int lane = threadIdx.x & 31;
    if (gw >= e) return;
    int s = esrc[gw];
    int d = edst[gw];
    float ns = nsrc[s];
    float4 v = ((const float4*)(x + (long)s * DFEAT))[lane];
    float* dst = agg + (long)d * DFEAT + lane * 4;
    atomicAdd(dst + 0, v.x * ns);
    atomicAdd(dst + 1, v.y * ns);
    atomicAdd(dst + 2, v.z * ns);
    atomicAdd(dst + 3, v.w * ns);
}

// ---------------- fused: hpre = prelu((agg*ndst) @ W + b, ac), + BN partial sums ----
// 256 threads = 8 waves; each wave -> 16 rows x 128 cols via 8 WMMA tiles, K=128 in steps of 4.
// hpre may alias agg (each wave reads all its K before writing its rows).
__global__ __launch_bounds__(256) void k_gemm_bn(const float* agg, float* hpre,
                                                 const float* __restrict__ W,
                                                 const float* __restrict__ bias,
                                                 const float* __restrict__ alpha_p,
                                                 const float* __restrict__ ndst,
                                                 float* __restrict__ fsum,
                                                 float* __restrict__ fsq, int n) {
    __shared__ float sW[16 * PSTR];  // 16 k-pairs (32 K-rows) of W, fragment-swizzled
    __shared__ float sSum[DFEAT];
    __shared__ float sSq[DFEAT];

    const int tid = threadIdx.x;
    if (tid < DFEAT) { sSum[tid] = 0.0f; sSq[tid] = 0.0f; }

    const int wave = tid >> 5;
    const int lane = tid & 31;
    const int half = lane >> 4;     // lane group 0/1
    const int mcol = lane & 15;     // A: row-in-tile; B/C: col-in-tile
    const int r0 = blockIdx.x * 128 + wave * 16;
    const int rowA = r0 + mcol;
    const bool aok = rowA < n;
    // nd==0 for OOB rows, so clamped loads contribute 0 -> no predication in the loop
    const float nd = aok ? ndst[rowA] : 0.0f;
    const int rowC = aok ? rowA : (n - 1);
    const float alpha = alpha_p[0];
    const float* aptr = agg + (long)rowC * DFEAT;

    v8f acc[8] = {};

    for (int kc = 0; kc < DFEAT; kc += 32) {
        __syncthreads();
        // stage 32 K-rows of W into LDS in fragment layout:
        // thread -> k-pair kp (0..15), 8 columns nb..nb+7; write (W[k][n],W[k+1][n]) pairs
        {
            int kp = tid >> 4;            // 0..15
            int nb = (tid & 15) * 8;      // 0,8,...,120
            const float* w0 = W + (kc + 2 * kp) * DFEAT + nb;
            const float* w1 = w0 + DFEAT;
            float4 r0a = ((const float4*)w0)[0];
            float4 r0b = ((const float4*)w0)[1];
            float4 r1a = ((const float4*)w1)[0];
            float4 r1b = ((const float4*)w1)[1];
            float2* dp = (float2*)(sW + kp * PSTR + nb * 2);
            dp[0] = make_float2(r0a.x, r1a.x);
            dp[1] = make_float2(r0a.y, r1a.y);
            dp[2] = make_float2(r0a.z, r1a.z);
            dp[3] = make_float2(r0a.w, r1a.w);
            dp[4] = make_float2(r0b.x, r1b.x);
            dp[5] = make_float2(r0b.y, r1b.y);
            dp[6] = make_float2(r0b.z, r1b.z);
            dp[7] = make_float2(r0b.w, r1b.w);
        }
        __syncthreads();

        for (int kb = 0; kb < 32; kb += 4) {
            const int kA = kc + kb + half * 2;   // global K for A
            float2 av = *(const float2*)(aptr + kA);
            v2f a;
            a.x = av.x * nd;
            a.y = av.y * nd;
            const float* bp = sW + (kb / 2 + half) * PSTR + mcol * 2;
#pragma unroll
            for (int t = 0; t < 8; ++t) {
                v2f b = *(const v2f*)(bp + t * 32);
                acc[t] = __builtin_amdgcn_wmma_f32_16x16x4_f32(
                    false, a, false, b, (short)0, acc[t], false, false);
            }
        }
    }

    // epilogue: bias + conv PReLU, store, BN partial sums
#pragma unroll
    for (int t = 0; t < 8; ++t) {
        const int col = t * 16 + mcol;
        const float bv = bias[col];
#pragma unroll
        for (int r = 0; r < 8; ++r) {
            const int row = r0 + half * 8 + r;
            if (row < n) {
                float v = acc[t][r] + bv;
                v = (v >= 0.0f) ? v : alpha * v;
                hpre[(long)row * DFEAT + col] = v;
                atomicAdd(&sSum[col], v);
                atomicAdd(&sSq[col], v * v);
            }
        }
    }
    __syncthreads();
    if (tid < DFEAT) {
        atomicAdd(&fsum[tid], sSum[tid]);
        atomicAdd(&fsq[tid], sSq[tid]);
    }
}

// ---------------- BN stats -> affine (a, c) so that y = a*h + c ----------------
__global__ void k_bnstats(const float* __restrict__ fsum, const float* __restrict__ fsq,
                          const float* __restrict__ gamma, const float* __restrict__ beta,
                          float* __restrict__ bnA, float* __restrict__ bnC, int n) {
    int c = threadIdx.x;
    float invn = 1.0f / (float)n;
    float mean = fsum[c] * invn;
    float var = fsq[c] * invn - mean * mean;
    float inv = rsqrtf(var + 1e-5f);
    float a = inv * gamma[c];
    bnA[c] = a;
    bnC[c] = beta[c] - mean * a;
}

// ---------------- BN apply + outer PReLU + graph segment-sum ----------------
// one float4 per thread over N*128 elements
__global__ __launch_bounds__(256) void k_bnapply(const float* __restrict__ hpre,
                                                 const float* __restrict__ bnA,
                                                 const float* __restrict__ bnC,
                                                 const float* __restrict__ alpha_p,
                                                 const int* __restrict__ gids,
                                                 float* __restrict__ hout,
                                                 float* __restrict__ gh, int n) {
    int idx = blockIdx.x * blockDim.x + threadIdx.x;  // float4 index
    int total = n * (DFEAT / 4);
    if (idx >= total) return;
    int node = idx >> 5;        // 32 float4 per row
    int q = idx & 31;
    float alpha = alpha_p[0];
    float4 v = ((const float4*)hpre)[idx];
    float4 a = ((const float4*)bnA)[q];
    float4 c = ((const float4*)bnC)[q];
    float4 y;
    y.x = fmaf(v.x, a.x, c.x); y.x = (y.x >= 0.0f) ? y.x : alpha * y.x;
    y.y = fmaf(v.y, a.y, c.y); y.y = (y.y >= 0.0f) ? y.y : alpha * y.y;
    y.z = fmaf(v.z, a.z, c.z); y.z = (y.z >= 0.0f) ? y.z : alpha * y.z;
    y.w = fmaf(v.w, a.w, c.w); y.w = (y.w >= 0.0f) ? y.w : alpha * y.w;
    ((float4*)hout)[idx] = y;
    int g = gids[node];
    float* gp = gh + (long)g * DFEAT + q * 4;
    atomicAdd(gp + 0, y.x);
    atomicAdd(gp + 1, y.y);
    atomicAdd(gp + 2, y.z);
    atomicAdd(gp + 3, y.w);
}

static inline int cdiv(long a, long b) { return (int)((a + b - 1) / b); }

extern "C" void kernel_launch(void* const* d_in, const int* in_sizes, int n_in,
                              void* d_out, int out_size, void* d_ws, size_t ws_size,
                              hipStream_t stream) {
    (void)n_in; (void)ws_size;
    const float* heat = (const float*)d_in[0];
    const int* esrc   = (const int*)d_in[1];
    const int* edst   = (const int*)d_in[2];
    const int* gids   = (const int*)d_in[3];
    const float* W0 = (const float*)d_in[4];
    const float* b0 = (const float*)d_in[5];
    const float* ac0 = (const float*)d_in[6];
    const float* g0 = (const float*)d_in[7];
    const float* be0 = (const float*)d_in[8];
    const float* aa0 = (const float*)d_in[9];
    const float* W1 = (const float*)d_in[10];
    const float* b1 = (const float*)d_in[11];
    const float* ac1 = (const float*)d_in[12];
    const float* g1 = (const float*)d_in[13];
    const float* be1 = (const float*)d_in[14];
    const float* aa1 = (const float*)d_in[15];

    const int E = in_sizes[1];
    const int N = in_sizes[3];
    const int ND = N * DFEAT;
    const int GD = out_size - ND;  // G * 128

    float* ws   = (float*)d_ws;
    float* nsrc = ws;                 // N  (deg_out -> norm_src)
    float* ndst = ws + N;             // N  (deg_in  -> norm_dst)
    float* agg  = ws + 2 * (long)N;   // N*128, also hpre in place
    float* h0   = agg + ND;           // N*128
    float* fsum = h0 + ND;            // 128
    float* fsq  = fsum + DFEAT;       // 128
    float* bnA  = fsq + DFEAT;        // 128
    float* bnC  = bnA + DFEAT;        // 128

    float* hout = (float*)d_out;  // N*128
    float* gh   = hout + ND;      // G*128

    const int gemm_blocks = cdiv(N, 128);

    // init
    k_zero<<<cdiv(2 * (long)N, 256), 256, 0, stream>>>(nsrc, 2 * N);
    k_zero<<<cdiv(ND, 256), 256, 0, stream>>>(agg, ND);
    k_zero<<<1, 256, 0, stream>>>(fsum, 2 * DFEAT);
    k_zero<<<cdiv(GD, 256), 256, 0, stream>>>(gh, GD);

    // degree norms
    k_deg<<<cdiv(E, 256), 256, 0, stream>>>(esrc, edst, nsrc, ndst, E);
    k_norm<<<cdiv(2 * (long)N, 256), 256, 0, stream>>>(nsrc, 2 * N);

    // ---- layer 0 ----
    k_scatter<<<cdiv((long)E * 32, 256), 256, 0, stream>>>(heat, nsrc, esrc, edst, agg, E);
    k_gemm_bn<<<gemm_blocks, 256, 0, stream>>>(agg, agg, W0, b0, ac0, ndst, fsum, fsq, N);
    k_bnstats<<<1, DFEAT, 0, stream>>>(fsum, fsq, g0, be0, bnA, bnC, N);
    k_bnapply<<<cdiv((long)N * 32, 256), 256, 0, stream>>>(agg, bnA, bnC, aa0, gids, h0, gh, N);

    // ---- layer 1 ----
    k_zero<<<cdiv(ND, 256), 256, 0, stream>>>(agg, ND);
    k_zero<<<1, 256, 0, stream>>>(fsum, 2 * DFEAT);
    k_scatter<<<cdiv((long)E * 32, 256), 256, 0, stream>>>(h0, nsrc, esrc, edst, agg, E);
    k_gemm_bn<<<gemm_blocks, 256, 0, stream>>>(agg, agg, W1, b1, ac1, ndst, fsum, fsq, N);
    k_bnstats<<<1, DFEAT, 0, stream>>>(fsum, fsq, g1, be1, bnA, bnC, N);
    k_bnapply<<<cdiv((long)N * 32, 256), 256, 0, stream>>>(agg, bnA, bnC, aa1, gids, hout, gh, N);
}